// GNNLayer_60876866453744
// MI455X (gfx1250) — compile-verified
//
#include <hip/hip_runtime.h>
#include <hip/hip_bf16.h>
#include <stdint.h>

typedef __attribute__((ext_vector_type(2))) float v2f;
typedef __attribute__((ext_vector_type(8))) float v8f;

#define Bdim 4
#define Ndim 10000
#define Fdim 128
#define Udim 128
#define Edim 160000

// ---------------------------------------------------------------- zero fill
__global__ __launch_bounds__(256) void zero_f32(float* __restrict__ p, int n) {
    int i = blockIdx.x * blockDim.x + threadIdx.x;
    int stride = gridDim.x * blockDim.x;
    for (; i < n; i += stride) p[i] = 0.0f;
}

// ------------------------------------------------- async W-chunk prefetch
// Copies a contiguous 32KB W chunk (64 K-rows x 128 cols fp32) into LDS via
// the CDNA5 async global->LDS path (no VGPR round trip, tracked by ASYNCcnt).
// Each of the 256 threads moves 8 x 16B. INST_OFFSET applies to both LDS and
// global addresses (CDNA5 ISA), and src/dst strides are equal,
// but we fold the stride into the addresses to keep the asm trivial.
__device__ __forceinline__ void async_w_prefetch(const float* __restrict__ Wsrc,
                                                 float* lds_dst, int tid)
{
    const float* src = Wsrc + tid * 4;                          // float4 #tid
    uint32_t dst = (uint32_t)(uintptr_t)(lds_dst + tid * 4);    // LDS byte off
#pragma unroll
    for (int i = 0; i < 8; ++i) {
        asm volatile("global_load_async_to_lds_b128 %0, %1, off"
                     :: "v"(dst), "v"(src) : "memory");
        dst += 256 * 16;    // 256 float4 per sweep = 4096 B
        src += 1024;        // 1024 floats
    }
}

__device__ __forceinline__ void wait_async_zero() {
    asm volatile("s_wait_asynccnt 0x0" ::: "memory");
}

// ---------------------------------------------------------------- WMMA GEMM
// out[M x 128] = relu( A[M x Ktot] * W[Ktot x 128] + bias ), fp32 WMMA.
// A is the virtual concat of A0 (K0 cols) and A1 (Ktot-K0 cols); K chunks of
// 64 always fall entirely inside one source (K0 is a multiple of 64).
// 8 waves / block, each wave owns a 16-row x 128-col output tile.
// W chunks are double-buffered in LDS and prefetched asynchronously.
__global__ __launch_bounds__(256) void gemm_relu_wmma(
    const float* __restrict__ A0, const float* __restrict__ A1,
    int K0, int Ktot,
    const float* __restrict__ W, const float* __restrict__ bias,
    float* __restrict__ out, int Mtiles)
{
    __shared__ float lds_w[2][64 * 128];           // 2 x 32 KB W chunks
    const int tid   = threadIdx.x;
    const int wave  = tid >> 5;
    const int lane  = tid & 31;
    const int khalf = lane >> 4;                   // A/B K-half per ISA layout
    const int nn    = lane & 15;

    int tile = blockIdx.x * 8 + wave;
    if (tile >= Mtiles) tile = Mtiles - 1;         // duplicate tile: keeps EXEC
    const int row0 = tile * 16;                    // all-ones for WMMA (ISA req)

    const v8f zero = {0.f,0.f,0.f,0.f,0.f,0.f,0.f,0.f};
    v8f acc[8];
#pragma unroll
    for (int t = 0; t < 8; ++t) acc[t] = zero;

    const int K1  = Ktot - K0;
    const int nch = Ktot / 64;

    // prefetch chunk 0
    async_w_prefetch(W, lds_w[0], tid);

    for (int c = 0; c < nch; ++c) {
        wait_async_zero();                         // own async stores landed
        __syncthreads();                           // chunk c visible block-wide

        if (c + 1 < nch)                           // overlap copy of c+1 with
            async_w_prefetch(W + (size_t)(c + 1) * 64 * 128,   // compute of c
                             lds_w[(c + 1) & 1], tid);

        const float* lw = lds_w[c & 1];
        const int kbase = c * 64;

        // A fragment base for this lane (row = nn, K offset = kbase + 2*khalf)
        const float* Arow;
        if (kbase < K0)
            Arow = A0 + (size_t)(row0 + nn) * K0 + kbase + 2 * khalf;
        else
            Arow = A1 + (size_t)(row0 + nn) * K1 + (kbase - K0) + 2 * khalf;

#pragma unroll
        for (int kb = 0; kb < 64; kb += 4) {
            v2f afrag;
            afrag.x = Arow[kb];
            afrag.y = Arow[kb + 1];
#pragma unroll
            for (int nt = 0; nt < 8; ++nt) {
                v2f bfrag;
                bfrag.x = lw[(kb + 2 * khalf    ) * 128 + nt * 16 + nn];
                bfrag.y = lw[(kb + 2 * khalf + 1) * 128 + nt * 16 + nn];
                acc[nt] = __builtin_amdgcn_wmma_f32_16x16x4_f32(
                    false, afrag, false, bfrag, (short)0, acc[nt], false, false);
            }
        }
        __syncthreads();   // all waves done reading buf (c&1) before reuse
    }

    // C/D layout: VGPR v -> M = v (lanes 0-15) / v+8 (lanes 16-31), N = lane%16
    const int mofs = khalf * 8;
#pragma unroll
    for (int nt = 0; nt < 8; ++nt) {
        const int col = nt * 16 + nn;
        const float bv = bias[col];
#pragma unroll
        for (int v = 0; v < 8; ++v) {
            float x = acc[nt][v] + bv;
            out[(size_t)(row0 + mofs + v) * 128 + col] = x > 0.0f ? x : 0.0f;
        }
    }
}

// ------------------------------------------------- edge attention + exp-sum
// One wave per (b,e): lanes split the 2*U=256 dot, wave32 xor-reduce,
// store exp(tanh(.)), accumulate global softmax denominator.
__global__ __launch_bounds__(256) void edge_att(
    const float* __restrict__ h, const long long* __restrict__ ei,
    const float* __restrict__ Wa, const float* __restrict__ ba,
    float* __restrict__ expv, float* __restrict__ sum)
{
    const int lane = threadIdx.x & 31;
    const int gw = (blockIdx.x * blockDim.x + threadIdx.x) >> 5;
    const int nw = (gridDim.x * blockDim.x) >> 5;
    const int u0 = lane * 4;
    const float4 wlo = *(const float4*)(Wa + u0);
    const float4 whi = *(const float4*)(Wa + 128 + u0);
    const float bav = ba[0];
    float lsum = 0.0f;
    const int P = Bdim * Edim;
    for (int p = gw; p < P; p += nw) {
        const int b = p / Edim, e = p - b * Edim;
        const long long s = ei[2 * e], t = ei[2 * e + 1];
        const float4 hs = *(const float4*)(h + ((size_t)b * Ndim + s) * Udim + u0);
        const float4 ht = *(const float4*)(h + ((size_t)b * Ndim + t) * Udim + u0);
        float part = hs.x*wlo.x + hs.y*wlo.y + hs.z*wlo.z + hs.w*wlo.w
                   + ht.x*whi.x + ht.y*whi.y + ht.z*whi.z + ht.w*whi.w;
#pragma unroll
        for (int m = 16; m >= 1; m >>= 1) part += __shfl_xor(part, m, 32);
        const float ex = __expf(tanhf(part + bav));   // tanh in (-1,1): no max needed
        if (lane == 0) { expv[p] = ex; lsum += ex; }
    }
    if (lane == 0)
        __hip_atomic_fetch_add(sum, lsum, __ATOMIC_RELAXED,
                               __HIP_MEMORY_SCOPE_AGENT);
}

// ------------------------------------------------------- weighted scatter-add
// One wave per (b,e): scattered[b,src,:] += attn * h[b,tgt,:] via L2 atomics.
__global__ __launch_bounds__(256) void edge_scatter(
    const float* __restrict__ h, const long long* __restrict__ ei,
    const float* __restrict__ expv, const float* __restrict__ sum,
    float* __restrict__ scattered)
{
    const int lane = threadIdx.x & 31;
    const int gw = (blockIdx.x * blockDim.x + threadIdx.x) >> 5;
    const int nw = (gridDim.x * blockDim.x) >> 5;
    const float inv = 1.0f / sum[0];
    const int u0 = lane * 4;
    const int P = Bdim * Edim;
    for (int p = gw; p < P; p += nw) {
        const int b = p / Edim, e = p - b * Edim;
        const long long s = ei[2 * e], t = ei[2 * e + 1];
        const float a = expv[p] * inv;
        const float4 ht = *(const float4*)(h + ((size_t)b * Ndim + t) * Udim + u0);
        float* dst = scattered + ((size_t)b * Ndim + s) * Udim + u0;
        __hip_atomic_fetch_add(dst + 0, a * ht.x, __ATOMIC_RELAXED, __HIP_MEMORY_SCOPE_AGENT);
        __hip_atomic_fetch_add(dst + 1, a * ht.y, __ATOMIC_RELAXED, __HIP_MEMORY_SCOPE_AGENT);
        __hip_atomic_fetch_add(dst + 2, a * ht.z, __ATOMIC_RELAXED, __HIP_MEMORY_SCOPE_AGENT);
        __hip_atomic_fetch_add(dst + 3, a * ht.w, __ATOMIC_RELAXED, __HIP_MEMORY_SCOPE_AGENT);
    }
}

// ---------------------------------------------------------------- launcher
extern "C" void kernel_launch(void* const* d_in, const int* in_sizes, int n_in,
                              void* d_out, int out_size, void* d_ws, size_t ws_size,
                              hipStream_t stream) {
    const float*     X  = (const float*)d_in[0];      // (B,N,F) fp32
    const long long* EI = (const long long*)d_in[1];  // (E,2) int64
    const float*     Wt = (const float*)d_in[2];      // (F,U)
    const float*     bt = (const float*)d_in[3];      // (U,)
    const float*     Wa = (const float*)d_in[4];      // (2U,1)
    const float*     ba = (const float*)d_in[5];      // (1,)
    const float*     Wc = (const float*)d_in[6];      // (2U,U)
    const float*     bc = (const float*)d_in[7];      // (U,)
    float* out = (float*)d_out;                       // (B,N,U) fp32

    float* ws = (float*)d_ws;
    const size_t HN = (size_t)Bdim * Ndim * Udim;     // 5,120,000 floats
    float* h   = ws;                                  // hidden features
    float* sc  = ws + HN;                             // scattered accumulator
    float* ev  = sc + HN;                             // exp(att) per (b,e)
    float* smm = ev + (size_t)Bdim * Edim;            // softmax denominator

    zero_f32<<<2048, 256, 0, stream>>>(sc, (int)HN);
    zero_f32<<<1, 64, 0, stream>>>(smm, 1);

    const int Mtiles  = (Bdim * Ndim) / 16;           // 2500
    const int gblocks = (Mtiles + 7) / 8;             // 313

    // h = relu(X * Wt + bt)
    gemm_relu_wmma<<<gblocks, 256, 0, stream>>>(X, nullptr, 128, 128, Wt, bt, h, Mtiles);
    // attention logits + global exp-sum
    edge_att<<<1024, 256, 0, stream>>>(h, EI, Wa, ba, ev, smm);
    // scattered = segment_sum(attn * h[tgt])
    edge_scatter<<<4096, 256, 0, stream>>>(h, EI, ev, smm, sc);
    // out = relu([h | scattered] * Wc + bc)
    gemm_relu_wmma<<<gblocks, 256, 0, stream>>>(h, sc, 128, 256, Wc, bc, out, Mtiles);
}